// dgl_gcn_18047452578197
// MI455X (gfx1250) — compile-verified
//
#include <hip/hip_runtime.h>
#include <hip/hip_bf16.h>

typedef __attribute__((ext_vector_type(2))) float v2f;
typedef __attribute__((ext_vector_type(8))) float v8f;

#define NN 100000
#define EE 1600000

// ---------------------------------------------------------------- utilities
__global__ void zero_f32_kernel(float* __restrict__ p, int n) {
    int i = blockIdx.x * blockDim.x + threadIdx.x;
    if (i < n) p[i] = 0.0f;
}

// out_deg / in_deg accumulation via f32 atomics (global_atomic_add_f32)
__global__ void degree_kernel(const int* __restrict__ src, const int* __restrict__ dst,
                              float* __restrict__ outdeg, float* __restrict__ indeg) {
    int i = blockIdx.x * blockDim.x + threadIdx.x;
    if (i < EE) {
        atomicAdd(&outdeg[src[i]], 1.0f);
        atomicAdd(&indeg[dst[i]], 1.0f);
    }
}

// deg -> rsqrt(clip(deg, 1))  in place, applied to both norm arrays at once
__global__ void norm_kernel(float* __restrict__ p, int n) {
    int i = blockIdx.x * blockDim.x + threadIdx.x;
    if (i < n) {
        float d = p[i];
        d = d < 1.0f ? 1.0f : d;
        p[i] = rsqrtf(d);
    }
}

// ------------------------------------------------------- GEMM1 (256 -> 64)
// h1[16 rows x 64 cols] per block. Block = 128 threads = 4 wave32; wave w owns
// output columns [16w, 16w+16). A-tile (x rows, pre-scaled by out_norm) staged
// in LDS. K stepped by 4 with V_WMMA_F32_16X16X4_F32.
__global__ __launch_bounds__(128) void gemm1_kernel(const float* __restrict__ x,
                                                    const float* __restrict__ W,   // [256,64]
                                                    const float* __restrict__ onorm,
                                                    float* __restrict__ h) {       // [N,64]
    __shared__ float xs[16 * 256];
    const int row0 = blockIdx.x * 16;
    const int tid  = threadIdx.x;

    // stage 16x256 tile, scaled by out_norm: 1024 float4 slots / 128 threads
    for (int i = tid; i < 16 * 64; i += 128) {
        int r  = i >> 6;     // row (64 float4 per row)
        int c4 = i & 63;
        float4 v = ((const float4*)(x + (size_t)(row0 + r) * 256))[c4];
        float  s = onorm[row0 + r];
        v.x *= s; v.y *= s; v.z *= s; v.w *= s;
        ((float4*)(xs + r * 256))[c4] = v;
    }
    __syncthreads();

    const int wave = tid >> 5;       // wave32
    const int lane = tid & 31;
    const int half = lane >> 4;      // 0: lanes 0-15, 1: lanes 16-31
    const int ln   = lane & 15;
    const int col0 = wave * 16;

    v8f c = {};
    for (int k0 = 0; k0 < 256; k0 += 4) {
        // A 16x4: lane ln holds row M=ln; v0/v1 = K = k0+2*half (+0,+1)
        v2f a, b;
        const float* ap = xs + ln * 256 + k0 + 2 * half;
        a.x = ap[0];
        a.y = ap[1];
        // B 4x16: VGPR0 -> rows K = k0+2*half, VGPR1 -> +1; N = ln
        const float* bp = W + (size_t)(k0 + 2 * half) * 64 + col0 + ln;
        b.x = bp[0];
        b.y = bp[64];
        c = __builtin_amdgcn_wmma_f32_16x16x4_f32(false, a, false, b, (short)0, c,
                                                  false, false);
    }
    // C/D: VGPR r -> (M = r + 8*half, N = ln)
    float* out = h + (size_t)row0 * 64 + col0;
#pragma unroll
    for (int r = 0; r < 8; ++r)
        out[(size_t)(r + 8 * half) * 64 + ln] = c[r];
}

// --------------------------------------------------------- edge aggregation
// layer 1: 64 feats -> 16 float4 chunks per edge
__global__ void agg64_kernel(const float* __restrict__ h, const int* __restrict__ src,
                             const int* __restrict__ dst, float* __restrict__ agg) {
    int idx = blockIdx.x * blockDim.x + threadIdx.x;
    if (idx >= EE * 16) return;
    int e = idx >> 4;
    int c = (idx & 15) * 4;
    int s = src[e], d = dst[e];
    float4 v = *(const float4*)(h + (size_t)s * 64 + c);
    float* q = agg + (size_t)d * 64 + c;
    atomicAdd(q + 0, v.x);
    atomicAdd(q + 1, v.y);
    atomicAdd(q + 2, v.z);
    atomicAdd(q + 3, v.w);
}

// layer 2: 40 feats -> 10 float4 chunks per edge, accumulate into d_out
__global__ void agg40_kernel(const float* __restrict__ h, const int* __restrict__ src,
                             const int* __restrict__ dst, float* __restrict__ agg) {
    int idx = blockIdx.x * blockDim.x + threadIdx.x;
    if (idx >= EE * 10) return;
    int e = idx / 10;
    int c = (idx % 10) * 4;
    int s = src[e], d = dst[e];
    float4 v = *(const float4*)(h + (size_t)s * 40 + c);
    float* q = agg + (size_t)d * 40 + c;
    atomicAdd(q + 0, v.x);
    atomicAdd(q + 1, v.y);
    atomicAdd(q + 2, v.z);
    atomicAdd(q + 3, v.w);
}

// agg * in_norm + bias, in place; width = feats
template <int F>
__global__ void finish_kernel(float* __restrict__ a, const float* __restrict__ inorm,
                              const float* __restrict__ bias) {
    int idx = blockIdx.x * blockDim.x + threadIdx.x;
    if (idx >= NN * F) return;
    int i = idx / F;
    int f = idx % F;
    a[idx] = a[idx] * inorm[i] + bias[f];
}

// -------------------------------------------------------- GEMM2 (64 -> 40)
// Block = 96 threads = 3 wave32 (3 column tiles cover 40 padded to 48;
// stores masked to col < 40). x2 tile staged in LDS.
__global__ __launch_bounds__(96) void gemm2_kernel(const float* __restrict__ x2,  // [N,64]
                                                   const float* __restrict__ W2,  // [64,40]
                                                   float* __restrict__ h2) {      // [N,40]
    __shared__ float xs[16 * 64];
    const int row0 = blockIdx.x * 16;
    const int tid  = threadIdx.x;

    for (int i = tid; i < 16 * 16; i += 96) {     // 256 float4 slots
        int r  = i >> 4;
        int c4 = i & 15;
        ((float4*)(xs + r * 64))[c4] = ((const float4*)(x2 + (size_t)(row0 + r) * 64))[c4];
    }
    __syncthreads();

    const int wave = tid >> 5;
    const int lane = tid & 31;
    const int half = lane >> 4;
    const int ln   = lane & 15;
    const int col0 = wave * 16;
    const int col  = col0 + ln;
    const bool valid = (col < 40);

    v8f c = {};
    for (int k0 = 0; k0 < 64; k0 += 4) {
        v2f a, b;
        const float* ap = xs + ln * 64 + k0 + 2 * half;
        a.x = ap[0];
        a.y = ap[1];
        int kk = k0 + 2 * half;
        b.x = valid ? W2[(size_t)kk * 40 + col] : 0.0f;
        b.y = valid ? W2[(size_t)(kk + 1) * 40 + col] : 0.0f;
        c = __builtin_amdgcn_wmma_f32_16x16x4_f32(false, a, false, b, (short)0, c,
                                                  false, false);
    }
    if (valid) {
        float* out = h2 + (size_t)row0 * 40 + col;
#pragma unroll
        for (int r = 0; r < 8; ++r)
            out[(size_t)(r + 8 * half) * 40] = c[r];
    }
}

// ------------------------------------------------------------------- launch
extern "C" void kernel_launch(void* const* d_in, const int* in_sizes, int n_in,
                              void* d_out, int out_size, void* d_ws, size_t ws_size,
                              hipStream_t stream) {
    const float* x   = (const float*)d_in[0];   // [N,256]
    const int*   src = (const int*)d_in[1];     // [E]
    const int*   dst = (const int*)d_in[2];     // [E]
    const float* W1  = (const float*)d_in[3];   // [256,64]
    const float* b1  = (const float*)d_in[4];   // [64]
    const float* W2  = (const float*)d_in[5];   // [64,40]
    const float* b2  = (const float*)d_in[6];   // [40]
    float* out = (float*)d_out;                 // [N,40]

    float* ws    = (float*)d_ws;
    float* onorm = ws;                          // [N]   (also holds out_deg first)
    float* inorm = ws + (size_t)NN;             // [N]
    float* h1    = ws + (size_t)2 * NN;         // [N,64]
    float* x2    = ws + (size_t)66 * NN;        // [N,64] agg1 then transformed in place
    float* h2    = ws + (size_t)130 * NN;       // [N,40]
    // total ws use: 170*N floats = 68 MB

    const int T = 256;
    // 1. zero deg/norm arrays, agg1 buffer, and the output (atomic target)
    zero_f32_kernel<<<(2 * NN + T - 1) / T, T, 0, stream>>>(onorm, 2 * NN);
    zero_f32_kernel<<<(64 * NN + T - 1) / T, T, 0, stream>>>(x2, 64 * NN);
    zero_f32_kernel<<<(40 * NN + T - 1) / T, T, 0, stream>>>(out, 40 * NN);

    // 2. degrees + norms
    degree_kernel<<<(EE + T - 1) / T, T, 0, stream>>>(src, dst, onorm, inorm);
    norm_kernel<<<(2 * NN + T - 1) / T, T, 0, stream>>>(onorm, 2 * NN);

    // 3. layer 1: h1 = (x * onorm) @ W1
    gemm1_kernel<<<NN / 16, 128, 0, stream>>>(x, W1, onorm, h1);
    agg64_kernel<<<(EE * 16 + T - 1) / T, T, 0, stream>>>(h1, src, dst, x2);
    finish_kernel<64><<<(NN * 64 + T - 1) / T, T, 0, stream>>>(x2, inorm, b1);

    // 4. layer 2: h2 = x2 @ W2
    gemm2_kernel<<<NN / 16, 96, 0, stream>>>(x2, W2, h2);
    agg40_kernel<<<(EE * 10 + T - 1) / T, T, 0, stream>>>(h2, src, dst, out);
    finish_kernel<40><<<(NN * 40 + T - 1) / T, T, 0, stream>>>(out, inorm, b2);
}